// LinearEncoder_14061722927346
// MI455X (gfx1250) — compile-verified
//
#include <hip/hip_runtime.h>

typedef __attribute__((ext_vector_type(2))) float v2f;
typedef __attribute__((ext_vector_type(8))) float v8f;

#define CH 128   // IN_C == OUT_C == 128

// ---------- degree / normalization ----------
__global__ void k_deg_init(float* __restrict__ deg, int n) {
  int i = blockIdx.x * blockDim.x + threadIdx.x;
  if (i < n) deg[i] = 1.0f;   // self loop contributes 1 to every target degree
}

__global__ void k_deg_count(const int* __restrict__ dst, float* __restrict__ deg, int e) {
  int i = blockIdx.x * blockDim.x + threadIdx.x;
  if (i < e) atomicAdd(&deg[dst[i]], 1.0f);
}

__global__ void k_deg_rsqrt(float* __restrict__ deg, int n) {
  int i = blockIdx.x * blockDim.x + threadIdx.x;
  if (i < n) deg[i] = rsqrtf(deg[i]);   // deg >= 1 always (self loop)
}

// ---------- h = x @ W via V_WMMA_F32_16X16X4_F32 ----------
// One wave -> one 16-row tile, all 128 output columns (8 N-tiles x 32 K-steps).
// A 16x4 layout : lanes 0-15 (M=lane, K=0/1), lanes 16-31 (M=lane-16, K=2/3)
// B 4x16 layout : lanes 0-15 (N=lane, K=0/1), lanes 16-31 (N=lane-16, K=2/3)
// C/D 16x16     : VGPR r -> M = r + 8*(lane>>4), N = lane&15
__global__ __launch_bounds__(128) void k_gemm_wmma(const float* __restrict__ x,
                                                   const float* __restrict__ W,
                                                   float* __restrict__ h,
                                                   int nrows) {
  const int lane = threadIdx.x & 31;
  const int wave = threadIdx.x >> 5;
  const int row0 = (blockIdx.x * 4 + wave) * 16;
  if (row0 >= nrows) return;             // wave-uniform: EXEC stays all-1s

  const int m  = lane & 15;
  const int kh = (lane >> 4) << 1;       // 0 or 2

  v8f acc[8];
#pragma unroll
  for (int t = 0; t < 8; ++t) acc[t] = (v8f){0.f,0.f,0.f,0.f,0.f,0.f,0.f,0.f};

  const float* xr = x + (size_t)(row0 + m) * CH;
#pragma unroll 4
  for (int k0 = 0; k0 < CH; k0 += 4) {
    v2f a;
    a[0] = xr[k0 + kh + 0];
    a[1] = xr[k0 + kh + 1];
    const float* w0 = W + (size_t)(k0 + kh) * CH + m;   // row k0+kh
#pragma unroll
    for (int t = 0; t < 8; ++t) {
      v2f b;
      b[0] = w0[t * 16];          // W[k0+kh  ][t*16 + m]
      b[1] = w0[t * 16 + CH];     // W[k0+kh+1][t*16 + m]
      acc[t] = __builtin_amdgcn_wmma_f32_16x16x4_f32(
          /*neg_a=*/false, a, /*neg_b=*/false, b,
          /*c_mod=*/(short)0, acc[t], /*reuse_a=*/false, /*reuse_b=*/false);
    }
  }

  const int rbase = row0 + ((lane >> 4) << 3);   // M = r + 8*(lane>>4)
#pragma unroll
  for (int t = 0; t < 8; ++t) {
#pragma unroll
    for (int r = 0; r < 8; ++r) {
      h[(size_t)(rbase + r) * CH + t * 16 + m] = acc[t][r];
    }
  }
}

// ---------- out = h * dinv^2 (self loop) + bias ----------
__global__ void k_out_init(const float* __restrict__ h, const float* __restrict__ dinv,
                           const float* __restrict__ bias, float* __restrict__ out, int n) {
  int i = blockIdx.x * blockDim.x + threadIdx.x;
  if (i < n * CH) {
    int node = i >> 7;
    int c    = i & (CH - 1);
    float di = dinv[node];
    out[i] = h[i] * di * di + bias[c];
  }
}

// ---------- edge aggregation: out[dst] += h[src] * dinv[src]*dinv[dst] ----------
// One wave per edge: 32 lanes x float4 = one coalesced 512B gather, then f32 atomics.
__global__ void k_scatter(const float* __restrict__ h, const float* __restrict__ dinv,
                          const int* __restrict__ src, const int* __restrict__ dst,
                          float* __restrict__ out, int e) {
  int wid  = (blockIdx.x * blockDim.x + threadIdx.x) >> 5;
  int lane = threadIdx.x & 31;
  if (wid >= e) return;
  int s = src[wid];
  int d = dst[wid];
  float norm = dinv[s] * dinv[d];
  const float4 hv = ((const float4*)(h + (size_t)s * CH))[lane];  // global_load_b128
  float* o = out + (size_t)d * CH + lane * 4;
  atomicAdd(o + 0, hv.x * norm);
  atomicAdd(o + 1, hv.y * norm);
  atomicAdd(o + 2, hv.z * norm);
  atomicAdd(o + 3, hv.w * norm);
}

extern "C" void kernel_launch(void* const* d_in, const int* in_sizes, int n_in,
                              void* d_out, int out_size, void* d_ws, size_t ws_size,
                              hipStream_t stream) {
  const float* x  = (const float*)d_in[0];
  const int*   ei = (const int*)d_in[1];   // [2,E] flat: first E = src, next E = dst
  const float* W  = (const float*)d_in[2];
  const float* bi = (const float*)d_in[3];
  float* out = (float*)d_out;

  const int n = in_sizes[0] / CH;
  const int e = in_sizes[1] / 2;
  const int* src = ei;
  const int* dst = ei + e;

  // workspace: h [n*CH f32] | deg->dinv [n f32]  (every byte rewritten each call)
  float* h   = (float*)d_ws;
  float* deg = h + (size_t)n * CH;

  k_deg_init <<<(n + 255) / 256, 256, 0, stream>>>(deg, n);
  k_deg_count<<<(e + 255) / 256, 256, 0, stream>>>(dst, deg, e);
  k_deg_rsqrt<<<(n + 255) / 256, 256, 0, stream>>>(deg, n);

  const int ntiles = (n + 15) / 16;                 // n = 100000 -> 6250 exact tiles
  k_gemm_wmma<<<(ntiles + 3) / 4, 128, 0, stream>>>(x, W, h, n);

  const int tot = n * CH;
  k_out_init<<<(tot + 255) / 256, 256, 0, stream>>>(h, deg, bi, out, n);

  // one wave (32 lanes) per edge
  long long threads = (long long)e * 32;
  k_scatter<<<(int)((threads + 255) / 256), 256, 0, stream>>>(h, deg, src, dst, out, e);
}